// MultiLayerGCN_43559558316604
// MI455X (gfx1250) — compile-verified
//
#include <hip/hip_runtime.h>
#include <hip/hip_bf16.h>

// MI455X (gfx1250) 3-layer GCN + mean-pool + linear head.
// Bottleneck is the edge scatter (L2-resident, ~2GB of gather+atomic traffic);
// GEMMs are done with fp32 WMMA (V_WMMA_F32_16X16X4_F32) since compute is free
// and fp32 matches reference numerics.

typedef __attribute__((ext_vector_type(2))) float v2f;
typedef __attribute__((ext_vector_type(8))) float v8f;

#define GCN_N 50000
#define GCN_E 1250000
#define GCN_H 64
#define GCN_G 64

// ---------------------------------------------------------------- zero init
__global__ void gcn_zero(float* __restrict__ deg, float* __restrict__ sums,
                         float* __restrict__ counts, int n) {
  int i = blockIdx.x * blockDim.x + threadIdx.x;
  if (i < n) deg[i] = 0.0f;
  if (i < GCN_G * GCN_H) sums[i] = 0.0f;
  if (i < GCN_G) counts[i] = 0.0f;
}

// ---------------------------------------------------------------- degree
__global__ void gcn_degree(const int* __restrict__ dst, float* __restrict__ deg,
                           int nEdges) {
  int e = blockIdx.x * blockDim.x + threadIdx.x;
  if (e < nEdges) unsafeAtomicAdd(&deg[dst[e]], 1.0f);
}

// deg -> dinv in place (self-loop adds 1, so never zero)
__global__ void gcn_dinv(float* __restrict__ degdinv, int n) {
  int i = blockIdx.x * blockDim.x + threadIdx.x;
  if (i < n) degdinv[i] = rsqrtf(degdinv[i] + 1.0f);
}

// ---------------------------------------------------------------- WMMA GEMM
// One wave computes a 16x16 tile of xw = relu?(h) @ W.
// Epilogue fuses self-loop + bias: agg = xw * dinv[row]^2 + bias[col].
// A (16x4 f32, 2 VGPRs): lanes 0-15 hold K=0,1; lanes 16-31 hold K=2,3.
// B (4x16 f32, 2 VGPRs): rows striped across lanes, same K split.
// C/D (16x16 f32, 8 VGPRs): VGPR v = row v (lanes 0-15) / row v+8 (lanes 16-31).
__global__ void __launch_bounds__(256)
gcn_gemm_wmma(const float* __restrict__ h, const float* __restrict__ W,
              const float* __restrict__ bias, const float* __restrict__ dinv,
              float* __restrict__ xw, float* __restrict__ agg,
              int nRowTiles, int applyRelu) {
  const int lane = threadIdx.x & 31;
  const int wave = threadIdx.x >> 5;
  const int rowTile = blockIdx.x * 2 + (wave >> 2);  // 2 row tiles per block
  if (rowTile >= nRowTiles) return;                  // wave-uniform branch
  const int m0 = rowTile * 16;
  const int n0 = (wave & 3) * 16;                    // 4 col tiles cover H=64
  const int half = lane >> 4;                        // 0: lanes 0-15, 1: 16-31
  const int l = lane & 15;
  const int arow = m0 + l;

  v8f c = {};
  #pragma unroll
  for (int k0 = 0; k0 < GCN_H; k0 += 4) {
    const int ka = k0 + 2 * half;
    float a0 = h[arow * GCN_H + ka];
    float a1 = h[arow * GCN_H + ka + 1];
    if (applyRelu) { a0 = fmaxf(a0, 0.0f); a1 = fmaxf(a1, 0.0f); }
    v2f a = {a0, a1};
    v2f b = {W[ka * GCN_H + n0 + l], W[(ka + 1) * GCN_H + n0 + l]};
    // 8 args: (neg_a, A, neg_b, B, c_mod, C, reuse_a, reuse_b)
    c = __builtin_amdgcn_wmma_f32_16x16x4_f32(false, a, false, b, (short)0, c,
                                              false, false);
  }

  const float bv = bias[n0 + l];
  #pragma unroll
  for (int v = 0; v < 8; ++v) {
    const int r = m0 + half * 8 + v;
    const float di = dinv[r];
    const float val = c[v];
    xw[r * GCN_H + n0 + l] = val;
    agg[r * GCN_H + n0 + l] = val * di * di + bv;  // self-loop + bias
  }
}

// ---------------------------------------------------------------- edge scatter
// 16 threads per edge, 4 channels each (one float4 load, 4 hw float atomics).
__global__ void gcn_scatter(const float* __restrict__ xw,
                            const int* __restrict__ src,
                            const int* __restrict__ dst,
                            const float* __restrict__ dinv,
                            float* __restrict__ agg, int nEdges) {
  int t = blockIdx.x * blockDim.x + threadIdx.x;
  int e = t >> 4;
  if (e >= nEdges) return;
  int q = t & 15;
  int s = src[e];
  int d = dst[e];
  float norm = dinv[s] * dinv[d];
  float4 v = ((const float4*)(xw + (size_t)s * GCN_H))[q];
  float* a = agg + (size_t)d * GCN_H + q * 4;
  unsafeAtomicAdd(a + 0, norm * v.x);
  unsafeAtomicAdd(a + 1, norm * v.y);
  unsafeAtomicAdd(a + 2, norm * v.z);
  unsafeAtomicAdd(a + 3, norm * v.w);
}

// ---------------------------------------------------------------- pooling
__global__ void gcn_pool(const float* __restrict__ h, const int* __restrict__ batch,
                         float* __restrict__ sums, float* __restrict__ counts,
                         int nNodes) {
  int t = blockIdx.x * blockDim.x + threadIdx.x;
  int n = t >> 4;
  if (n >= nNodes) return;
  int q = t & 15;
  int g = batch[n];
  float4 v = ((const float4*)(h + (size_t)n * GCN_H))[q];
  float* s = sums + g * GCN_H + q * 4;
  unsafeAtomicAdd(s + 0, v.x);
  unsafeAtomicAdd(s + 1, v.y);
  unsafeAtomicAdd(s + 2, v.z);
  unsafeAtomicAdd(s + 3, v.w);
  if (q == 0) unsafeAtomicAdd(&counts[g], 1.0f);
}

// ---------------------------------------------------------------- head
__global__ void gcn_head(const float* __restrict__ sums,
                         const float* __restrict__ counts,
                         const float* __restrict__ Wp, const float* __restrict__ bp,
                         float* __restrict__ out) {
  int g = threadIdx.x;
  if (g >= GCN_G) return;
  float cinv = 1.0f / fmaxf(counts[g], 1.0f);
  float acc = 0.0f;
  #pragma unroll
  for (int j = 0; j < GCN_H; ++j) acc += sums[g * GCN_H + j] * Wp[j];
  out[g] = acc * cinv + bp[0];
}

// ---------------------------------------------------------------- launch
extern "C" void kernel_launch(void* const* d_in, const int* in_sizes, int n_in,
                              void* d_out, int out_size, void* d_ws, size_t ws_size,
                              hipStream_t stream) {
  const float* x   = (const float*)d_in[0];
  const int*   ei  = (const int*)d_in[1];   // [2,E] flat: src then dst
  const int*   bat = (const int*)d_in[2];
  const float* W0  = (const float*)d_in[3];
  const float* b0  = (const float*)d_in[4];
  const float* W1  = (const float*)d_in[5];
  const float* b1  = (const float*)d_in[6];
  const float* W2  = (const float*)d_in[7];
  const float* b2  = (const float*)d_in[8];
  const float* Wp  = (const float*)d_in[9];
  const float* bp  = (const float*)d_in[10];
  float* out = (float*)d_out;

  const int N = in_sizes[0] / GCN_H;     // 50000
  const int E = in_sizes[1] / 2;         // 1250000
  const int Npad = ((N + 63) / 64) * 64; // keep float4 alignment downstream

  // Workspace layout (floats)
  float* ws   = (float*)d_ws;
  float* dinv = ws;                      // N (deg, then rsqrt in place)
  float* xw   = dinv + Npad;             // N*H
  float* bufA = xw + (size_t)Npad * GCN_H;
  float* bufB = bufA + (size_t)Npad * GCN_H;
  float* sums = bufB + (size_t)Npad * GCN_H; // G*H
  float* cnts = sums + GCN_G * GCN_H;        // G

  const int* src = ei;
  const int* dst = ei + E;

  const int T = 256;
  const int nRowTiles = (N + 15) / 16;
  dim3 gemmGrid((nRowTiles + 1) / 2);
  dim3 scatGrid(((size_t)E * 16 + T - 1) / T);
  dim3 poolGrid(((size_t)N * 16 + T - 1) / T);

  // degree -> dinv (once; graph is fixed across layers)
  gcn_zero<<<(N + T - 1) / T, T, 0, stream>>>(dinv, sums, cnts, N);
  gcn_degree<<<(E + T - 1) / T, T, 0, stream>>>(dst, dinv, E);
  gcn_dinv<<<(N + T - 1) / T, T, 0, stream>>>(dinv, N);

  // layer 0: x -> bufA
  gcn_gemm_wmma<<<gemmGrid, T, 0, stream>>>(x, W0, b0, dinv, xw, bufA, nRowTiles, 0);
  gcn_scatter<<<scatGrid, T, 0, stream>>>(xw, src, dst, dinv, bufA, E);

  // layer 1: relu(bufA) -> bufB
  gcn_gemm_wmma<<<gemmGrid, T, 0, stream>>>(bufA, W1, b1, dinv, xw, bufB, nRowTiles, 1);
  gcn_scatter<<<scatGrid, T, 0, stream>>>(xw, src, dst, dinv, bufB, E);

  // layer 2: relu(bufB) -> bufA
  gcn_gemm_wmma<<<gemmGrid, T, 0, stream>>>(bufB, W2, b2, dinv, xw, bufA, nRowTiles, 1);
  gcn_scatter<<<scatGrid, T, 0, stream>>>(xw, src, dst, dinv, bufA, E);

  // pool + head
  gcn_pool<<<poolGrid, T, 0, stream>>>(bufA, bat, sums, cnts, N);
  gcn_head<<<1, 64, 0, stream>>>(sums, cnts, Wp, bp, out);
}